// DSQGAttentionV8J12_9500467659261
// MI455X (gfx1250) — compile-verified
//
#include <hip/hip_runtime.h>
#include <hip/hip_bf16.h>
#include <stdint.h>

typedef __attribute__((ext_vector_type(2))) float v2f;
typedef __attribute__((ext_vector_type(8))) float v8f;
typedef int v4i __attribute__((vector_size(16)));   // matches builtin prototype

#define B_    2
#define H_    16
#define N_    4096
#define HD_   64
#define J_    12
#define TROWS 16
#define QSTR  68   // padded LDS row stride in floats: 272B, 16B aligned, conflict-friendly

// ---- CDNA5 async global->LDS copy (ASYNCcnt path) -------------------------
__device__ __forceinline__ void async_ld_b128(uint32_t lds_off, const float* gaddr) {
#if __has_builtin(__builtin_amdgcn_global_load_async_to_lds_b128)
  __builtin_amdgcn_global_load_async_to_lds_b128(
      (__attribute__((address_space(1))) v4i*)(uintptr_t)gaddr,
      (__attribute__((address_space(3))) v4i*)(uintptr_t)lds_off,
      /*imm offset*/0, /*cpol*/0);
#else
  asm volatile("global_load_async_to_lds_b128 %0, %1, off"
               :: "v"(lds_off), "v"(gaddr) : "memory");
#endif
}

__device__ __forceinline__ void wait_async0() {
#if __has_builtin(__builtin_amdgcn_s_wait_asynccnt)
  __builtin_amdgcn_s_wait_asynccnt(0);
#else
  asm volatile("s_wait_asynccnt 0x0" ::: "memory");
#endif
}

// One wave32 per (b, h, 16-row tile).
__global__ __launch_bounds__(32)
void dsqg_attn_kernel(const float* __restrict__ q, const float* __restrict__ k,
                      const float* __restrict__ v, const float* __restrict__ pb,
                      const float* __restrict__ se,
                      const float* __restrict__ phase_base,
                      const float* __restrict__ phase_gain,
                      const float* __restrict__ y_pre,
                      const float* __restrict__ z_pre,
                      float* __restrict__ out) {
  constexpr int OFF[J_] = {1, 2, 4, 8, 16, 64, 96, 192, 384, 512, 768, 1024};

  __shared__ __align__(16) float q_s[TROWS * QSTR];        //  4.3 KB
  __shared__ __align__(16) float k_s[J_ * TROWS * QSTR];   // 52.2 KB
  __shared__ __align__(16) float se_s[16 * HD_];           //  4.0 KB (rows 12..15 zero)
  __shared__ __align__(16) float d_s[16 * 16];             //  1.0 KB

  const int lane  = threadIdx.x;   // 0..31
  const int lrow  = lane & 15;     // tile row this lane owns
  const int lhalf = lane >> 4;     // which 32-column half of HD

  const int bid  = blockIdx.x;
  const int tile = bid & (N_ / TROWS - 1);
  const int head = (bid >> 8) & (H_ - 1);
  const int b    = bid >> 12;
  const int t    = tile * TROWS;
  const size_t bhN = ((size_t)b * H_ + head) * (size_t)N_;

  const float* qbase = q + bhN * HD_;
  const float* kbase = k + bhN * HD_;
  const float* vbase = v + bhN * HD_;

  const uint32_t lds_q = (uint32_t)(uintptr_t)&q_s[0];
  const uint32_t lds_k = (uint32_t)(uintptr_t)&k_s[0];

  // ---- async stage Q tile (16x64 f32): 8 b128 issues, 512B each -----------
  #pragma unroll
  for (int j = 0; j < 8; ++j) {
    int idx = j * 32 + lane;          // 16B chunk id 0..255
    int r = idx >> 4, c = idx & 15;
    async_ld_b128(lds_q + (uint32_t)(r * QSTR + c * 4) * 4u,
                  qbase + (size_t)(t + r) * HD_ + c * 4);
  }
  // ---- async stage 12 shifted K tiles (rows clamped; masked later) --------
  #pragma unroll
  for (int i = 0; i < J_; ++i) {
    #pragma unroll
    for (int j = 0; j < 8; ++j) {
      int idx = j * 32 + lane;
      int r = idx >> 4, c = idx & 15;
      int sr = t + r - OFF[i]; sr = sr < 0 ? 0 : sr;
      async_ld_b128(lds_k + (uint32_t)((i * TROWS + r) * QSTR + c * 4) * 4u,
                    kbase + (size_t)sr * HD_ + c * 4);
    }
  }
  // ---- tiny SE matrix into LDS (zero-padded to 16 rows for WMMA B) --------
  for (int idx = lane; idx < 16 * HD_; idx += 32)
    se_s[idx] = (idx < J_ * HD_) ? se[idx] : 0.0f;

  wait_async0();
  __syncthreads();

  // ---- preload this lane's q half-row into registers ----------------------
  float qreg[32];
  #pragma unroll
  for (int c4 = 0; c4 < 8; ++c4) {
    float4 t4 = *(const float4*)&q_s[lrow * QSTR + lhalf * 32 + c4 * 4];
    qreg[c4 * 4 + 0] = t4.x; qreg[c4 * 4 + 1] = t4.y;
    qreg[c4 * 4 + 2] = t4.z; qreg[c4 * 4 + 3] = t4.w;
  }

  // ---- diagonal q . k_{n-d} dots on VALU (half-row partial + xor16) -------
  float dots[J_];
  #pragma unroll
  for (int i = 0; i < J_; ++i) {
    float p = 0.0f;
    #pragma unroll
    for (int c4 = 0; c4 < 8; ++c4) {
      float4 kv = *(const float4*)&k_s[(i * TROWS + lrow) * QSTR + lhalf * 32 + c4 * 4];
      p = fmaf(qreg[c4 * 4 + 0], kv.x, p);
      p = fmaf(qreg[c4 * 4 + 1], kv.y, p);
      p = fmaf(qreg[c4 * 4 + 2], kv.z, p);
      p = fmaf(qreg[c4 * 4 + 3], kv.w, p);
    }
    p += __shfl_xor(p, 16, 32);   // combine column halves; both halves get full dot
    dots[i] = p;
  }

  // ---- q . se_i for all 12 offsets via V_WMMA_F32_16X16X4_F32 -------------
  // D(16x16) = Q_tile(16x64) x SE^T(64x16), K chained in 16 steps of 4.
  // A frag: lanes 0-15 K={4k,4k+1}, lanes 16-31 K={4k+2,4k+3} (ISA 16x4 layout).
  // B frag mirrors: vgpr0 K=4k(+2*half), vgpr1 K=4k+1(+2*half), column = lane%16.
  v8f acc = {0.f, 0.f, 0.f, 0.f, 0.f, 0.f, 0.f, 0.f};
  #pragma unroll
  for (int kk = 0; kk < 16; ++kk) {
    const int kb = 4 * kk + 2 * lhalf;
    v2f af, bf2;
    af.x  = q_s[lrow * QSTR + kb + 0];
    af.y  = q_s[lrow * QSTR + kb + 1];
    bf2.x = se_s[lrow * HD_ + kb + 0];   // B[k][n] = se[n][k], column n = lrow
    bf2.y = se_s[lrow * HD_ + kb + 1];
    acc = __builtin_amdgcn_wmma_f32_16x16x4_f32(false, af, false, bf2,
                                                (short)0, acc, false, false);
  }
  // transpose WMMA C/D layout -> per-row layout through LDS
  #pragma unroll
  for (int r = 0; r < 8; ++r)
    d_s[(r + 8 * lhalf) * 16 + lrow] = acc[r];
  __syncthreads();

  // ---- scores, causal mask, softmax over the 12 offsets -------------------
  const int n = t + lrow;
  const float sc = 0.125f;  // 1/sqrt(64)
  float sscore[J_];
  #pragma unroll
  for (int i = 0; i < J_; ++i) {
    float s = (dots[i] + d_s[lrow * 16 + i]) * sc + pb[i * H_ + head];
    sscore[i] = (n >= OFF[i]) ? s : -__builtin_inff();
  }
  float mx = -__builtin_inff();
  #pragma unroll
  for (int i = 0; i < J_; ++i) mx = fmaxf(mx, sscore[i]);
  float alpha[J_];
  float ssum = 0.0f;
  #pragma unroll
  for (int i = 0; i < J_; ++i) {
    float e = (n >= OFF[i]) ? __expf(sscore[i] - mx) : 0.0f;
    alpha[i] = e; ssum += e;
  }
  const float inv = (ssum > 0.0f) ? (1.0f / ssum) : 0.0f;
  #pragma unroll
  for (int i = 0; i < J_; ++i) alpha[i] *= inv;

  // ---- alpha-weighted (phase-rotated) value accumulation ------------------
  const float yv0 = y_pre[(bhN + n) * 2 + 0];
  const float yv1 = y_pre[(bhN + n) * 2 + 1];
  float o[32];
  #pragma unroll
  for (int c = 0; c < 32; ++c) o[c] = 0.0f;

  #pragma unroll
  for (int i = 0; i < J_; ++i) {
    int sr = n - OFF[i];
    int cr = sr < 0 ? 0 : sr;             // clamp; alpha==0 masks the garbage
    const float a = alpha[i];
    const float* vrow = vbase + (size_t)cr * HD_ + lhalf * 32;
    float4 vv[8];
    #pragma unroll
    for (int c4 = 0; c4 < 8; ++c4) vv[c4] = ((const float4*)vrow)[c4];
    if (i >= 4 && lhalf == 0) {           // rotation touches dims 0..3 only
      const int pi = i - 4;
      const float b0 = phase_base[(pi * H_ + head) * 2 + 0];
      const float b1 = phase_base[(pi * H_ + head) * 2 + 1];
      const float g0 = phase_gain[(pi * H_ + head) * 2 + 0];
      const float g1 = phase_gain[(pi * H_ + head) * 2 + 1];
      const float z0 = z_pre[(bhN + cr) * 2 + 0];
      const float z1 = z_pre[(bhN + cr) * 2 + 1];
      const float th0 = b0 + g0 * yv0 * z0;
      const float th1 = b1 + g1 * yv1 * z1;
      const float c0 = __cosf(th0), s0 = __sinf(th0);
      const float c1 = __cosf(th1), s1 = __sinf(th1);
      const float r0 = c0 * vv[0].x - s0 * vv[0].y;
      const float r1 = s0 * vv[0].x + c0 * vv[0].y;
      const float r2 = c1 * vv[0].z - s1 * vv[0].w;
      const float r3 = s1 * vv[0].z + c1 * vv[0].w;
      vv[0].x = r0; vv[0].y = r1; vv[0].z = r2; vv[0].w = r3;
    }
    #pragma unroll
    for (int c4 = 0; c4 < 8; ++c4) {
      o[c4 * 4 + 0] = fmaf(a, vv[c4].x, o[c4 * 4 + 0]);
      o[c4 * 4 + 1] = fmaf(a, vv[c4].y, o[c4 * 4 + 1]);
      o[c4 * 4 + 2] = fmaf(a, vv[c4].z, o[c4 * 4 + 2]);
      o[c4 * 4 + 3] = fmaf(a, vv[c4].w, o[c4 * 4 + 3]);
    }
  }

  float* orow = out + (bhN + n) * HD_ + lhalf * 32;
  #pragma unroll
  for (int c4 = 0; c4 < 8; ++c4) {
    ((float4*)orow)[c4] = make_float4(o[c4 * 4 + 0], o[c4 * 4 + 1],
                                      o[c4 * 4 + 2], o[c4 * 4 + 3]);
  }
}

extern "C" void kernel_launch(void* const* d_in, const int* in_sizes, int n_in,
                              void* d_out, int out_size, void* d_ws, size_t ws_size,
                              hipStream_t stream) {
  const float* q          = (const float*)d_in[0];
  const float* k          = (const float*)d_in[1];
  const float* v          = (const float*)d_in[2];
  const float* pb         = (const float*)d_in[3];
  const float* se         = (const float*)d_in[4];
  const float* phase_base = (const float*)d_in[5];
  const float* phase_gain = (const float*)d_in[6];
  const float* y_pre      = (const float*)d_in[7];
  const float* z_pre      = (const float*)d_in[8];
  float* out              = (float*)d_out;

  const int blocks = B_ * H_ * (N_ / TROWS);  // 8192 wave32 blocks
  dsqg_attn_kernel<<<blocks, 32, 0, stream>>>(q, k, v, pb, se, phase_base,
                                              phase_gain, y_pre, z_pre, out);
}